// MLPPredictor_9869834846314
// MI455X (gfx1250) — compile-verified
//
#include <hip/hip_runtime.h>
#include <hip/hip_bf16.h>

// ---------------------------------------------------------------------------
// Problem: score[e,o] = Wu[o].x[src[e]] + Wv[o].x[dst[e]] + b[o]
//   N_NODES=10000, N_EDGES=640000, D=128, O=2
// Strategy:
//   Phase 0: build zero-padded transposed B table Bpad[16][128] (8 KB) in ws.
//   Phase 1 (WMMA f32 16x16x4): P[node,0:2] = x[node].Wu^T, P[node,2:4] = x[node].Wv^T
//            — branch-free inner loop, uniform b64 loads for A and B.
//   Phase 2 (bandwidth): out[e] = P[src[e],0:2] + P[dst[e],2:4] + b
// ---------------------------------------------------------------------------

typedef __attribute__((ext_vector_type(2))) float v2f;
typedef __attribute__((ext_vector_type(8))) float v8f;

#define D_FEAT   128
#define N_NODES  10000
#define N_TILES  (N_NODES / 16)     // 625, exact
#define WAVES_PER_BLOCK 8

// Phase 0: Bpad[n][k] for n in 0..15, k in 0..127.
//   n=0 -> Wu[0], n=1 -> Wu[1], n=2 -> Wv[0], n=3 -> Wv[1], n>=4 -> 0.
__global__ __launch_bounds__(256)
void build_bpad(const float* __restrict__ W,   // (2, 256) row-major
                float* __restrict__ Bpad)      // (16, 128)
{
    const int i = (int)blockIdx.x * 256 + (int)threadIdx.x;   // 0..2047
    if (i >= 16 * 128) return;
    const int n = i >> 7;
    const int k = i & 127;
    float v = 0.0f;
    if (n < 4) v = W[(size_t)(n & 1) * 256 + (size_t)(n >> 1) * 128 + k];
    Bpad[i] = v;
}

// Phase 1: one wave32 computes a 16-row tile of P via V_WMMA_F32_16X16X4_F32.
// A (16x4 f32): lanes 0-15 -> M=lane, K = {2*half, 2*half+1}, half = lane>>4.
// B (4x16 f32): mirrored — lane's pair comes from Bpad[n][k+2*half .. +1].
__global__ __launch_bounds__(WAVES_PER_BLOCK * 32)
void node_project_wmma(const float* __restrict__ x,
                       const float* __restrict__ Bpad,  // (16, 128) padded
                       float* __restrict__ P)           // (N_NODES, 4)
{
    const int wave = (int)threadIdx.x >> 5;
    const int lane = (int)threadIdx.x & 31;
    const int tile = (int)blockIdx.x * WAVES_PER_BLOCK + wave;
    if (tile >= N_TILES) return;                        // wave-uniform branch

    const int half = lane >> 4;        // 0: lanes 0-15, 1: lanes 16-31
    const int m    = lane & 15;        // row within tile for A; col index for B/D
    const int row  = tile * 16 + m;

    const float* arow = x    + (size_t)row * D_FEAT + 2 * half;  // A pairs
    const float* brow = Bpad + (size_t)m   * D_FEAT + 2 * half;  // B pairs

    v8f c = {};
#pragma unroll
    for (int k = 0; k < D_FEAT; k += 4) {
        v2f a  = *(const v2f*)(arow + k);
        v2f bv = *(const v2f*)(brow + k);
        // 8 args: (neg_a, A, neg_b, B, c_mod, C, reuse_a, reuse_b)
        c = __builtin_amdgcn_wmma_f32_16x16x4_f32(
                false, a, false, bv, (short)0, c, false, false);
    }

    // D (16x16 f32, 8 VGPRs): VGPR r -> M = r + 8*half, N = m. Keep N<4 only.
    if (m < 4) {
        float* prow = P + (size_t)(tile * 16 + 8 * half) * 4 + m;
#pragma unroll
        for (int r = 0; r < 8; ++r)
            prow[(size_t)r * 4] = c[r];
    }
}

// Phase 2: 4 edges per thread. int4 index loads, float2 gathers from the
// 160KB P table (L2-resident), float4 coalesced stores.
__global__ __launch_bounds__(256)
void edge_score(const int*  __restrict__ src,
                const int*  __restrict__ dst,
                const float2* __restrict__ P2,   // P viewed as (N_NODES, 2) float2
                const float* __restrict__ b,
                float4* __restrict__ out4,       // out viewed as float4 (2 edges each)
                int nquads)
{
    const int t = (int)blockIdx.x * (int)blockDim.x + (int)threadIdx.x;
    if (t >= nquads) return;

    const int4 s = ((const int4*)src)[t];
    const int4 d = ((const int4*)dst)[t];
    const float b0 = b[0], b1 = b[1];

    const float2 u0 = P2[2 * s.x],     u1 = P2[2 * s.y];
    const float2 u2 = P2[2 * s.z],     u3 = P2[2 * s.w];
    const float2 w0 = P2[2 * d.x + 1], w1 = P2[2 * d.y + 1];
    const float2 w2 = P2[2 * d.z + 1], w3 = P2[2 * d.w + 1];

    float4 oA, oB;
    oA.x = u0.x + w0.x + b0;  oA.y = u0.y + w0.y + b1;
    oA.z = u1.x + w1.x + b0;  oA.w = u1.y + w1.y + b1;
    oB.x = u2.x + w2.x + b0;  oB.y = u2.y + w2.y + b1;
    oB.z = u3.x + w3.x + b0;  oB.w = u3.y + w3.y + b1;

    out4[2 * t + 0] = oA;
    out4[2 * t + 1] = oB;
}

extern "C" void kernel_launch(void* const* d_in, const int* in_sizes, int n_in,
                              void* d_out, int out_size, void* d_ws, size_t ws_size,
                              hipStream_t stream)
{
    const float* x   = (const float*)d_in[0];   // (10000, 128) f32
    const int*   src = (const int*)  d_in[1];   // (640000,) i32 (JAX x64 off)
    const int*   dst = (const int*)  d_in[2];   // (640000,) i32
    const float* W   = (const float*)d_in[3];   // (2, 256) f32
    const float* b   = (const float*)d_in[4];   // (2,) f32

    // Workspace layout: P (10000*4 f32 = 160000 B, 16B-aligned), then Bpad (8 KB).
    float* P    = (float*)d_ws;
    float* Bpad = (float*)((char*)d_ws + (size_t)N_NODES * 4 * sizeof(float));

    // Phase 0: 2048 elements -> 8 blocks of 256.
    build_bpad<<<8, 256, 0, stream>>>(W, Bpad);

    // Phase 1: 625 tiles, 8 waves/block -> 79 blocks of 256 threads.
    const int blocks1 = (N_TILES + WAVES_PER_BLOCK - 1) / WAVES_PER_BLOCK;
    node_project_wmma<<<blocks1, WAVES_PER_BLOCK * 32, 0, stream>>>(x, Bpad, P);

    // Phase 2: 640000 edges / 4 per thread = 160000 threads.
    const int n_edges = in_sizes[1];
    const int nquads  = n_edges / 4;
    const int blocks2 = (nquads + 255) / 256;
    edge_score<<<blocks2, 256, 0, stream>>>(src, dst, (const float2*)P, b,
                                            (float4*)d_out, nquads);
}